// CustomRenderer_81836306858695
// MI455X (gfx1250) — compile-verified
//
#include <hip/hip_runtime.h>
#include <cmath>

// CDNA5 / gfx1250, wave32. Matrix core formulation of point-to-segment distance:
//   e[M][N] = (p-v).s   via V_WMMA_F32_16X16X4_F32  (A row = [sx, sy, -v.s, 0],  B col = [px, py, 1, 0])
//   f[M][N] = |p-v|^2   via V_WMMA_F32_16X16X4_F32  (A row = [1, -2vx, -2vy, |v|^2], B col = [|p|^2, px, py, 1])
// then dist^2 = f - 2*t*e + t^2*d2 with t = clamp(e/(d2+eps), 0, 1).

typedef float v2f __attribute__((ext_vector_type(2)));
typedef float v8f __attribute__((ext_vector_type(8)));

#define NPTS   32
#define NSEG   (NPTS - 1)
#define EPSF   1e-5f
#define DARKMIN 1e-8f
#define BIGF   1e30f

__global__ __launch_bounds__(256) void traj_render_wmma(
    const float* __restrict__ traj,     // [B][NPTS][2]
    const float* __restrict__ radius_p, // [1]
    const float* __restrict__ exp_p,    // [1]
    const float* __restrict__ dx_p,     // [1]
    const float* __restrict__ dy_p,     // [1]
    float* __restrict__ out,            // [B][W][W]
    int W, int tilesPerBatch, int B)
{
    const int lane = threadIdx.x & 31;
    const int gw   = (int)((blockIdx.x * blockDim.x + threadIdx.x) >> 5);
    const int b    = gw / tilesPerBatch;
    if (b >= B) return;                       // wave-uniform: waves are whole
    const int tile = gw - b * tilesPerBatch;
    const int q    = lane & 15;
    const bool hi  = (lane >= 16);

    const float Wf   = (float)W;
    const float sdx  = dx_p[0];
    const float sdy  = dy_p[0];
    const float radius = radius_p[0];
    const float dexp   = exp_p[0];

    __builtin_prefetch(traj + b * (NPTS * 2), 0, 3);  // global_prefetch_b8

    // This wave's 16 pixels: pi0..pi0+15. Center the coordinate frame on the
    // tile so |p| is tiny and the |p|^2 - 2 v.p + |v|^2 expansion is stable
    // exactly where distances are small.
    const int   pi0 = tile * 16;
    const int   pi  = pi0 + q;
    const float ox  = (float)(pi0 % W) + 7.5f;
    const float oy  = (float)(pi0 / W);
    const float px  = (float)(pi % W) - ox;
    const float py  = (float)(pi / W) - oy;
    const float pp  = px * px + py * py;

    // B fragments (4x16, f32): V0 = K0 (lanes<16) / K2 (lanes>=16); V1 = K1 / K3.
    const v2f b_e = hi ? (v2f){1.f, 0.f} : (v2f){px, py};   // col = [px, py, 1, 0]
    const v2f b_f = hi ? (v2f){py, 1.f} : (v2f){pp, px};    // col = [pp, px, py, 1]

    const float2* tp    = (const float2*)traj;
    const int     tbase = b * NPTS;

    float mind2 = 3.0e38f;
    const int srcBase = hi ? 8 : 0;   // C/D rows this half-wave holds: M = srcBase + r

    #pragma unroll
    for (int c = 0; c < (NSEG + 15) / 16; ++c) {
        const int  m     = c * 16 + q;          // segment this lane describes
        const bool valid = (m < NSEG);

        float vx = 0.f, vy = 0.f, wx = 0.f, wy = 0.f;
        if (valid) {
            float2 v2 = tp[tbase + m];
            float2 w2 = tp[tbase + m + 1];
            vx = (v2.x + sdx) * Wf - ox;
            vy = (v2.y + sdy) * Wf - oy;
            wx = (w2.x + sdx) * Wf - ox;
            wy = (w2.y + sdy) * Wf - oy;
        }
        const float sx    = wx - vx;
        const float sy    = wy - vy;
        const float d2    = sx * sx + sy * sy;
        const float invd2 = 1.0f / (d2 + EPSF);
        const float vds   = vx * sx + vy * sy;
        const float vv    = valid ? (vx * vx + vy * vy) : BIGF;

        // A fragments (16x4, f32): lanes<16 hold K0,K1 of row M=lane;
        // lanes>=16 hold K2,K3 of row M=lane-16.
        // pad rows: e -> [0,0,0,0] (e=0, t=0), f -> [0,0,0,BIG] (dist^2=BIG)
        const v2f a_e = hi ? (v2f){-vds, 0.f} : (v2f){sx, sy};
        const v2f a_f = hi ? (v2f){valid ? -2.f * vy : 0.f, vv}
                           : (v2f){valid ? 1.f : 0.f, valid ? -2.f * vx : 0.f};

        v8f zero8 = {0.f, 0.f, 0.f, 0.f, 0.f, 0.f, 0.f, 0.f};
        // (neg_a, A, neg_b, B, c_mod, C, reuse_a, reuse_b)
        v8f e_m = __builtin_amdgcn_wmma_f32_16x16x4_f32(
            false, a_e, false, b_e, (short)0, zero8, false, false);
        v8f f_m = __builtin_amdgcn_wmma_f32_16x16x4_f32(
            false, a_f, false, b_f, (short)0, zero8, false, false);

        // Epilogue: this lane owns pixel N=q, rows M = srcBase..srcBase+7.
        // Segment M's d2/invd2 live in lane M (and M+16); grab via shuffle.
        #pragma unroll
        for (int r = 0; r < 8; ++r) {
            const float d2m  = __shfl(d2,    srcBase + r, 32);
            const float id2m = __shfl(invd2, srcBase + r, 32);
            const float e = e_m[r];
            const float f = f_m[r];
            float t = e * id2m;
            t = fminf(fmaxf(t, 0.f), 1.f);
            // dist^2 = f - 2*t*e + t^2*d2  (exact expansion of |p-(v+t*s)|^2)
            float dd = fmaf(t, fmaf(t, d2m, -(e + e)), f);
            dd = fmaxf(dd, 0.f);
            mind2 = fminf(mind2, dd);
        }
    }

    // merge the two half-wave row groups (M 0-7 vs 8-15): same pixel N lives
    // in lanes L and L+16.
    mind2 = fminf(mind2, __shfl_xor(mind2, 16, 32));

    if (!hi && pi < W * W) {
        const float dist = sqrtf(mind2);
        float d = (radius - dist) / radius;
        d = fminf(fmaxf(d, DARKMIN), 1.0f);
        float dark = __powf(d, dexp);          // exp2(dexp * log2(d)), d > 0
        dark = fminf(fmaxf(dark, 0.f), 1.f);
        out[b * W * W + pi] = dark;
    }
}

extern "C" void kernel_launch(void* const* d_in, const int* in_sizes, int n_in,
                              void* d_out, int out_size, void* d_ws, size_t ws_size,
                              hipStream_t stream) {
    const float* traj   = (const float*)d_in[0];
    const float* radius = (const float*)d_in[1];
    const float* dexp   = (const float*)d_in[2];
    const float* dx     = (const float*)d_in[3];
    const float* dy     = (const float*)d_in[4];
    float* out = (float*)d_out;

    int B = in_sizes[0] / (NPTS * 2);          // traj is [B][32][2]
    if (B < 1) B = 1;
    const int wpix = out_size / B;             // W*W
    const int W = (int)(sqrtf((float)wpix) + 0.5f);
    const int tilesPerBatch = (W * W + 15) / 16;

    const long long waves   = (long long)B * tilesPerBatch;
    const long long threads = waves * 32;      // wave32
    const int block = 256;                     // 8 waves / block
    const int grid  = (int)((threads + block - 1) / block);

    traj_render_wmma<<<grid, block, 0, stream>>>(traj, radius, dexp, dx, dy,
                                                 out, W, tilesPerBatch, B);
}